// Lambda_45707041964069
// MI455X (gfx1250) — compile-verified
//
#include <hip/hip_runtime.h>

// Smoluchowski step: dn = gain - n*(K@nm), nm = n with nm[0]=0, dn[0]=n[0].
// gain[k] = sum over anti-diagonal i+j=k (k<N) of K[i,j]*nm[i]*nm[j].
//
// Bandwidth-bound: K = 256 MB streamed once -> ~11us floor at 23.3 TB/s.
// CDNA5 paths used:
//  * global_load_async_to_lds_b128 + s_wait_asynccnt : async tile staging
//  * v_wmma_f32_16x16x4_f32 : loss GEMV row-sums on the matrix pipe
//    (B columns all equal a wcol 4-slice -> D[m][*] = rowsum partial)
//  * ds_add_f32 : conflict-free anti-diagonal binning for the gain term

#define TM 64          // tile rows
#define TN 128         // tile cols
#define LDST 132       // padded LDS row stride (floats): bank = (4r+c)%64,
                       // conflict-free A-fragment reads; 132*4=528 keeps 16B align
#define NTHREADS 256

typedef __attribute__((address_space(3))) float lds_f32;
typedef __attribute__((ext_vector_type(2))) float v2f;
typedef __attribute__((ext_vector_type(8))) float v8f;

__device__ __forceinline__ unsigned lds_byte_addr(const float* p) {
    // generic -> LDS address space; async/DS instructions take the raw LDS
    // byte offset (hardware adds the wave's LDS base).
    return (unsigned)(unsigned long long)(lds_f32*)(p);
}

__global__ __launch_bounds__(NTHREADS) void smol_zero_kernel(float* a, float* b, int N) {
    int i = blockIdx.x * blockDim.x + threadIdx.x;
    if (i < N) { a[i] = 0.0f; b[i] = 0.0f; }
}

__global__ __launch_bounds__(NTHREADS) void smol_tile_kernel(
    const float* __restrict__ n, const float* __restrict__ K,
    float* __restrict__ gain, float* __restrict__ rowsum, int N)
{
    __shared__ float ldsK[TM * LDST];
    __shared__ float wrow[TM];
    __shared__ float wcol[TN];
    __shared__ float bins[TM + TN - 1];

    const int tid = threadIdx.x;
    const int C = blockIdx.x * TN;   // tile column base
    const int R = blockIdx.y * TM;   // tile row base

    // ---- CDNA5 async DMA: stage the 64x128 f32 tile of K into LDS ----
    // 8192 floats = 256 threads * 8 * float4; 16B aligned, fully coalesced.
    {
        const float* gbase = K + (size_t)R * (size_t)N + (size_t)C;
        #pragma unroll
        for (int p = 0; p < 8; ++p) {
            const int e = p * (NTHREADS * 4) + tid * 4;   // float index in tile
            const int r = e >> 7;                         // / TN
            const int c = e & (TN - 1);
            unsigned lds_addr = lds_byte_addr(&ldsK[r * LDST + c]);
            unsigned long long gaddr =
                (unsigned long long)(gbase + (size_t)r * (size_t)N + (size_t)c);
            asm volatile("global_load_async_to_lds_b128 %0, %1, off"
                         :: "v"(lds_addr), "v"(gaddr)
                         : "memory");
        }
    }

    // Stage nm fragments + zero diagonal bins while the DMA is in flight.
    if (tid < TM) { int g = R + tid; wrow[tid] = (g == 0) ? 0.0f : n[g]; }
    if (tid < TN) { int g = C + tid; wcol[tid] = (g == 0) ? 0.0f : n[g]; }
    if (tid < TM + TN - 1) bins[tid] = 0.0f;

    asm volatile("s_wait_asynccnt 0" ::: "memory");
    __syncthreads();

    const int wave = tid >> 5;
    const int lane = tid & 31;

    // ================= loss GEMV on the WMMA pipe =================
    // wave -> row-block rb (16 rows) and column half h (64 cols).
    // D = sum_ch A(16x4) x B(4x16) + C, with B[k][*] = wcol[cbase+k]
    // => every lane's D column holds the rowsum partial for 16 rows.
    {
        const int rb   = wave >> 1;               // 0..3
        const int h    = wave & 1;                // 0..1
        const int arow = rb * 16 + (lane & 15);
        const int kk   = 2 * (lane >> 4);         // A/B K-offset per ISA layout

        v8f acc = {};
        #pragma unroll
        for (int ch = 0; ch < 16; ++ch) {
            const int cb = h * 64 + ch * 4 + kk;  // even -> 8B aligned
            v2f a = *(const v2f*)&ldsK[arow * LDST + cb];
            v2f b = *(const v2f*)&wcol[cb];
            acc = __builtin_amdgcn_wmma_f32_16x16x4_f32(
                      false, a, false, b, (short)0, acc, false, false);
        }
        // D layout: lane 0 holds rows rb*16+0..7, lane 16 holds rows rb*16+8..15
        if ((lane & 15) == 0) {
            const int rbase = R + rb * 16 + (lane >> 4) * 8;
            #pragma unroll
            for (int i = 0; i < 8; ++i)
                atomicAdd(&rowsum[rbase + i], acc[i]);
        }
    }

    // ================= gain anti-diagonal binning (VALU+LDS) =================
    // Tiles entirely above the anti-diagonal (R+C >= N) contribute nothing.
    if ((R + C) < N) {
        const float wc0 = wcol[lane],      wc1 = wcol[lane + 32],
                    wc2 = wcol[lane + 64], wc3 = wcol[lane + 96];
        #pragma unroll
        for (int rr = 0; rr < TM / 8; ++rr) {
            const int r = wave * (TM / 8) + rr;
            const float wr = wrow[r];
            const float k0 = ldsK[r * LDST + lane];
            const float k1 = ldsK[r * LDST + lane + 32];
            const float k2 = ldsK[r * LDST + lane + 64];
            const float k3 = ldsK[r * LDST + lane + 96];
            // 32 consecutive bin addresses per ds_add wave -> conflict-free.
            atomicAdd(&bins[r + lane      ], k0 * wc0 * wr);
            atomicAdd(&bins[r + lane + 32 ], k1 * wc1 * wr);
            atomicAdd(&bins[r + lane + 64 ], k2 * wc2 * wr);
            atomicAdd(&bins[r + lane + 96 ], k3 * wc3 * wr);
        }

        __syncthreads();

        // Flush tile diagonal bins to the global gain accumulator.
        if (tid < TM + TN - 1) {
            const int d = R + C + tid;
            if (d < N) atomicAdd(&gain[d], bins[tid]);
        }
    }
}

__global__ __launch_bounds__(NTHREADS) void smol_finalize_kernel(
    const float* __restrict__ n, const float* __restrict__ rowsum,
    float* __restrict__ out, int N)
{
    int i = blockIdx.x * blockDim.x + threadIdx.x;
    if (i < N) {
        float v = out[i] - n[i] * rowsum[i];   // gain - loss
        out[i] = (i == 0) ? n[0] : v;
    }
}

extern "C" void kernel_launch(void* const* d_in, const int* in_sizes, int n_in,
                              void* d_out, int out_size, void* d_ws, size_t ws_size,
                              hipStream_t stream) {
    // inputs: d_in[0]=t (scalar, unused), d_in[1]=n (N f32), d_in[2]=K (N*N f32)
    const float* n = (const float*)d_in[1];
    const float* K = (const float*)d_in[2];
    float* gain   = (float*)d_out;   // accumulates gain, then finalized in place
    float* rowsum = (float*)d_ws;    // N floats of scratch: sum_j K[i,j]*nm[j]
    const int N = in_sizes[1];

    const int vb = (N + NTHREADS - 1) / NTHREADS;
    smol_zero_kernel<<<vb, NTHREADS, 0, stream>>>(gain, rowsum, N);

    dim3 grid(N / TN, N / TM);
    smol_tile_kernel<<<grid, NTHREADS, 0, stream>>>(n, K, gain, rowsum, N);

    smol_finalize_kernel<<<vb, NTHREADS, 0, stream>>>(n, rowsum, gain, N);
}